// Filtration_6828998001463
// MI455X (gfx1250) — compile-verified
//
#include <hip/hip_runtime.h>
#include <math.h>

typedef float v2f __attribute__((ext_vector_type(2)));
typedef float v8f __attribute__((ext_vector_type(8)));

#define SLOPE 0.01f
#define BN_EPS 1e-5f

__device__ __forceinline__ float lrelu(float x) { return x > 0.0f ? x : SLOPE * x; }

// ---------------- scatter (segment_sum) kernels ----------------
__global__ void k_scatter3(const int* __restrict__ src, const int* __restrict__ dst,
                           const float* __restrict__ pos, float* __restrict__ agg, int E) {
  int e = blockIdx.x * blockDim.x + threadIdx.x;
  if (e >= E) return;
  int s = src[e] * 3, d = dst[e] * 3;
  unsafeAtomicAdd(&agg[d + 0], pos[s + 0]);
  unsafeAtomicAdd(&agg[d + 1], pos[s + 1]);
  unsafeAtomicAdd(&agg[d + 2], pos[s + 2]);
}

// 16 threads per edge, 4 features each (float4 gather, 4 f32 atomics)
__global__ void k_scatter64(const int* __restrict__ src, const int* __restrict__ dst,
                            const float* __restrict__ X, float* __restrict__ agg, int E) {
  int gid = blockIdx.x * blockDim.x + threadIdx.x;
  if (gid >= E * 16) return;
  int e  = gid >> 4;
  int fb = (gid & 15) << 2;
  int s = src[e] * 64 + fb;
  int d = dst[e] * 64 + fb;
  float4 v = *(const float4*)(X + s);
  unsafeAtomicAdd(&agg[d + 0], v.x);
  unsafeAtomicAdd(&agg[d + 1], v.y);
  unsafeAtomicAdd(&agg[d + 2], v.z);
  unsafeAtomicAdd(&agg[d + 3], v.w);
}

// ---------------- layer-1 node MLP (3 -> 3 -> 64), K too small for WMMA ----
__global__ void k_gin1_node(const float* __restrict__ pos, const float* __restrict__ agg1,
                            const float* __restrict__ eps1,
                            const float* __restrict__ W1a, const float* __restrict__ b1a,
                            const float* __restrict__ W1b, const float* __restrict__ b1b,
                            float* __restrict__ H1, int N) {
  int i = blockIdx.x * blockDim.x + threadIdx.x;
  if (i >= N) return;
  float e = 1.0f + eps1[0];
  float a0 = e * pos[i * 3 + 0] + agg1[i * 3 + 0];
  float a1 = e * pos[i * 3 + 1] + agg1[i * 3 + 1];
  float a2 = e * pos[i * 3 + 2] + agg1[i * 3 + 2];
  float u0 = lrelu(a0 * W1a[0] + a1 * W1a[3] + a2 * W1a[6] + b1a[0]);
  float u1 = lrelu(a0 * W1a[1] + a1 * W1a[4] + a2 * W1a[7] + b1a[1]);
  float u2 = lrelu(a0 * W1a[2] + a1 * W1a[5] + a2 * W1a[8] + b1a[2]);
  float* out = H1 + (size_t)i * 64;
  #pragma unroll 8
  for (int f = 0; f < 64; ++f)
    out[f] = u0 * W1b[f] + u1 * W1b[64 + f] + u2 * W1b[128 + f] + b1b[f];
}

// ---------------- BatchNorm: stats / finalize / apply ----------------
__global__ __launch_bounds__(256) void k_bn_stats(const float* __restrict__ X, int N,
                                                  float* __restrict__ gsum, float* __restrict__ gsq) {
  __shared__ float ss[256], sq[256];
  int tid = threadIdx.x;
  int f = tid & 63, rsub = tid >> 6;
  int chunk = (N + gridDim.x - 1) / gridDim.x;
  int r0 = blockIdx.x * chunk;
  int r1 = min(N, r0 + chunk);
  float a = 0.0f, b = 0.0f;
  for (int r = r0 + rsub; r < r1; r += 4) {
    float v = X[(size_t)r * 64 + f];
    a += v; b += v * v;
  }
  ss[tid] = a; sq[tid] = b;
  __syncthreads();
  if (tid < 64) {
    a = ss[tid] + ss[tid + 64] + ss[tid + 128] + ss[tid + 192];
    b = sq[tid] + sq[tid + 64] + sq[tid + 128] + sq[tid + 192];
    atomicAdd(&gsum[tid], a);
    atomicAdd(&gsq[tid], b);
  }
}

__global__ void k_bn_finalize(const float* __restrict__ gsum, const float* __restrict__ gsq,
                              const float* __restrict__ g, const float* __restrict__ b,
                              int N, float* __restrict__ scale, float* __restrict__ shift) {
  int f = threadIdx.x;
  if (f < 64) {
    float inv = 1.0f / (float)N;
    float mu  = gsum[f] * inv;
    float var = fmaxf(gsq[f] * inv - mu * mu, 0.0f);
    float sc  = g[f] * rsqrtf(var + BN_EPS);
    scale[f] = sc;
    shift[f] = b[f] - mu * sc;
  }
}

__global__ void k_bn_apply(const float* __restrict__ X, const float* __restrict__ scale,
                           const float* __restrict__ shift, float* __restrict__ Y, int total) {
  int i = blockIdx.x * blockDim.x + threadIdx.x;
  if (i >= total) return;
  int f = i & 63;
  Y[i] = lrelu(X[i] * scale[f] + shift[f]);
}

// ---------------- layer-2 MLP: fused (1+eps)x+agg -> GEMM -> LReLU -> GEMM --
// One wave = 16 output rows; WMMA f32 16x16x4; LDS row stride 65 avoids bank
// conflicts on the K-column reads for the second GEMM.
__global__ __launch_bounds__(256) void k_gin2_mlp(
    const float* __restrict__ X1, const float* __restrict__ agg,
    const float* __restrict__ eps2,
    const float* __restrict__ W2a, const float* __restrict__ b2a,
    const float* __restrict__ W2b, const float* __restrict__ b2b,
    float* __restrict__ H2, int N) {
  __shared__ float lds[8 * 16 * 65];
  const int tid  = threadIdx.x;
  const int wave = tid >> 5;
  const int lane = tid & 31;
  const int lr   = lane & 15;
  const int hi   = lane >> 4;
  const int koff = hi * 2;
  const int row0 = (blockIdx.x * 8 + wave) * 16;
  int row = row0 + lr;
  if (row >= N) row = N - 1;
  const float e2 = 1.0f + eps2[0];
  float* tileLds = lds + wave * (16 * 65);

  v8f acc[4] = {};
  #pragma unroll
  for (int kk = 0; kk < 64; kk += 4) {
    const int c0 = kk + koff;
    v2f a;
    a.x = e2 * X1[(size_t)row * 64 + c0]     + agg[(size_t)row * 64 + c0];
    a.y = e2 * X1[(size_t)row * 64 + c0 + 1] + agg[(size_t)row * 64 + c0 + 1];
    #pragma unroll
    for (int nt = 0; nt < 4; ++nt) {
      const int col = nt * 16 + lr;
      v2f b;
      b.x = W2a[(c0    ) * 64 + col];
      b.y = W2a[(c0 + 1) * 64 + col];
      acc[nt] = __builtin_amdgcn_wmma_f32_16x16x4_f32(false, a, false, b, (short)0,
                                                      acc[nt], false, false);
    }
  }
  #pragma unroll
  for (int v = 0; v < 8; ++v) {
    const int m = v + hi * 8;
    #pragma unroll
    for (int nt = 0; nt < 4; ++nt) {
      const int col = nt * 16 + lr;
      tileLds[m * 65 + col] = lrelu(acc[nt][v] + b2a[col]);
    }
  }
  __syncthreads();

  v8f acc2[4] = {};
  #pragma unroll
  for (int kk = 0; kk < 64; kk += 4) {
    const int c0 = kk + koff;
    v2f a;
    a.x = tileLds[lr * 65 + c0];
    a.y = tileLds[lr * 65 + c0 + 1];
    #pragma unroll
    for (int nt = 0; nt < 4; ++nt) {
      const int col = nt * 16 + lr;
      v2f b;
      b.x = W2b[(c0    ) * 64 + col];
      b.y = W2b[(c0 + 1) * 64 + col];
      acc2[nt] = __builtin_amdgcn_wmma_f32_16x16x4_f32(false, a, false, b, (short)0,
                                                       acc2[nt], false, false);
    }
  }
  #pragma unroll
  for (int v = 0; v < 8; ++v) {
    const int m = v + hi * 8;
    if (row0 + m < N) {
      #pragma unroll
      for (int nt = 0; nt < 4; ++nt) {
        const int col = nt * 16 + lr;
        H2[(size_t)(row0 + m) * 64 + col] = acc2[nt][v] + b2b[col];
      }
    }
  }
}

// ---------------- final GEMM: [emb_deg|emb_lab|x2](192) @ Wf1 + bf1 ----------
__global__ __launch_bounds__(256) void k_final_gemm(
    const int* __restrict__ node_deg, const int* __restrict__ node_lab,
    const float* __restrict__ emb_deg, const float* __restrict__ emb_lab,
    const float* __restrict__ X2,
    const float* __restrict__ Wf1, const float* __restrict__ bf1,
    float* __restrict__ ZF, int N) {
  const int tid  = threadIdx.x;
  const int wave = tid >> 5;
  const int lane = tid & 31;
  const int lr   = lane & 15;
  const int hi   = lane >> 4;
  const int koff = hi * 2;
  const int row0 = (blockIdx.x * 8 + wave) * 16;
  int row = row0 + lr;
  if (row >= N) row = N - 1;
  const int deg64 = node_deg[row] * 64;
  const int lab64 = node_lab[row] * 64;
  const size_t r64 = (size_t)row * 64;

  v8f acc[4] = {};
  for (int kk = 0; kk < 192; kk += 4) {
    const int c0 = kk + koff;
    v2f a;
    if (kk < 64) {
      a.x = emb_deg[deg64 + c0];
      a.y = emb_deg[deg64 + c0 + 1];
    } else if (kk < 128) {
      a.x = emb_lab[lab64 + c0 - 64];
      a.y = emb_lab[lab64 + c0 - 63];
    } else {
      a.x = X2[r64 + c0 - 128];
      a.y = X2[r64 + c0 - 127];
    }
    #pragma unroll
    for (int nt = 0; nt < 4; ++nt) {
      const int col = nt * 16 + lr;
      v2f b;
      b.x = Wf1[(c0    ) * 64 + col];
      b.y = Wf1[(c0 + 1) * 64 + col];
      acc[nt] = __builtin_amdgcn_wmma_f32_16x16x4_f32(false, a, false, b, (short)0,
                                                      acc[nt], false, false);
    }
  }
  #pragma unroll
  for (int v = 0; v < 8; ++v) {
    const int m = v + hi * 8;
    if (row0 + m < N) {
      #pragma unroll
      for (int nt = 0; nt < 4; ++nt) {
        const int col = nt * 16 + lr;
        ZF[(size_t)(row0 + m) * 64 + col] = acc[nt][v] + bf1[col];
      }
    }
  }
}

// ---------------- head: BN+LReLU fused into 64->1 dot + sigmoid --------------
__global__ void k_head(const float* __restrict__ ZF, const float* __restrict__ scale,
                       const float* __restrict__ shift, const float* __restrict__ Wf2,
                       const float* __restrict__ bf2, float* __restrict__ out, int N) {
  int i = blockIdx.x * blockDim.x + threadIdx.x;
  if (i >= N) return;
  const float* z = ZF + (size_t)i * 64;
  float acc = bf2[0];
  #pragma unroll 8
  for (int f = 0; f < 64; ++f)
    acc += lrelu(z[f] * scale[f] + shift[f]) * Wf2[f];
  out[i] = 1.0f / (1.0f + expf(-acc));
}

// ---------------------------------------------------------------------------
extern "C" void kernel_launch(void* const* d_in, const int* in_sizes, int n_in,
                              void* d_out, int out_size, void* d_ws, size_t ws_size,
                              hipStream_t stream) {
  const int*   node_deg = (const int*)d_in[0];
  const int*   node_lab = (const int*)d_in[1];
  const int*   edge     = (const int*)d_in[2];
  const float* pos      = (const float*)d_in[3];
  const float* emb_deg  = (const float*)d_in[4];
  const float* emb_lab  = (const float*)d_in[5];
  const float* eps1     = (const float*)d_in[6];
  const float* W1a      = (const float*)d_in[7];
  const float* b1a      = (const float*)d_in[8];
  const float* W1b      = (const float*)d_in[9];
  const float* b1b      = (const float*)d_in[10];
  const float* g1       = (const float*)d_in[11];
  const float* be1      = (const float*)d_in[12];
  const float* eps2     = (const float*)d_in[13];
  const float* W2a      = (const float*)d_in[14];
  const float* b2a      = (const float*)d_in[15];
  const float* W2b      = (const float*)d_in[16];
  const float* b2b      = (const float*)d_in[17];
  const float* g2       = (const float*)d_in[18];
  const float* be2      = (const float*)d_in[19];
  const float* Wf1      = (const float*)d_in[20];
  const float* bf1      = (const float*)d_in[21];
  const float* gf       = (const float*)d_in[22];
  const float* bef      = (const float*)d_in[23];
  const float* Wf2      = (const float*)d_in[24];
  const float* bf2      = (const float*)d_in[25];

  const int N = in_sizes[0];
  const int E = in_sizes[2] / 2;
  const int* src = edge;
  const int* dst = edge + E;

  float* ws = (float*)d_ws;
  const size_t n64 = (size_t)N * 64;
  float* bufA  = ws;               // H1 pre-BN, then H2 pre-BN
  float* bufB  = ws + n64;         // X1, then ZF pre-BN
  float* bufC  = ws + 2 * n64;     // agg2, then X2
  float* agg1  = ws + 3 * n64;     // N x 3
  float* stats = agg1 + (size_t)N * 3;   // 3 stages x [sum64|sq64|scale64|shift64]
  float* S0 = stats;
  float* S1 = stats + 256;
  float* S2 = stats + 512;

  // zero all accumulators (fresh every launch — replay safe)
  hipMemsetAsync(agg1, 0, (size_t)N * 3 * sizeof(float), stream);
  hipMemsetAsync(bufC, 0, n64 * sizeof(float), stream);
  hipMemsetAsync(stats, 0, 3 * 256 * sizeof(float), stream);

  const int tiles   = (N + 15) / 16;
  const int gBlocks = (tiles + 7) / 8;

  // ---- GIN layer 1 ----
  k_scatter3<<<(E + 255) / 256, 256, 0, stream>>>(src, dst, pos, agg1, E);
  k_gin1_node<<<(N + 255) / 256, 256, 0, stream>>>(pos, agg1, eps1, W1a, b1a, W1b, b1b, bufA, N);
  k_bn_stats<<<512, 256, 0, stream>>>(bufA, N, S0, S0 + 64);
  k_bn_finalize<<<1, 64, 0, stream>>>(S0, S0 + 64, g1, be1, N, S0 + 128, S0 + 192);
  k_bn_apply<<<(N * 64 + 255) / 256, 256, 0, stream>>>(bufA, S0 + 128, S0 + 192, bufB, N * 64);

  // ---- GIN layer 2 ----
  k_scatter64<<<(E * 16 + 255) / 256, 256, 0, stream>>>(src, dst, bufB, bufC, E);
  k_gin2_mlp<<<gBlocks, 256, 0, stream>>>(bufB, bufC, eps2, W2a, b2a, W2b, b2b, bufA, N);
  k_bn_stats<<<512, 256, 0, stream>>>(bufA, N, S1, S1 + 64);
  k_bn_finalize<<<1, 64, 0, stream>>>(S1, S1 + 64, g2, be2, N, S1 + 128, S1 + 192);
  k_bn_apply<<<(N * 64 + 255) / 256, 256, 0, stream>>>(bufA, S1 + 128, S1 + 192, bufC, N * 64);

  // ---- final MLP + head ----
  k_final_gemm<<<gBlocks, 256, 0, stream>>>(node_deg, node_lab, emb_deg, emb_lab,
                                            bufC, Wf1, bf1, bufB, N);
  k_bn_stats<<<512, 256, 0, stream>>>(bufB, N, S2, S2 + 64);
  k_bn_finalize<<<1, 64, 0, stream>>>(S2, S2 + 64, gf, bef, N, S2 + 128, S2 + 192);
  k_head<<<(N + 255) / 256, 256, 0, stream>>>(bufB, S2 + 128, S2 + 192, Wf2, bf2,
                                              (float*)d_out, N);
}